// GraphSAGE_Net_72060961292398
// MI455X (gfx1250) — compile-verified
//
#include <hip/hip_runtime.h>
#include <stdint.h>

typedef __attribute__((ext_vector_type(2))) float v2f;
typedef __attribute__((ext_vector_type(8))) float v8f;

#define NNODES 50000
#define NEDGES 800000
#define DIN  64
#define DHID 128
#define DOUT 64

// ---------------------------------------------------------------- zero fill
__global__ void k_zero(float* __restrict__ p, long n) {
    long i = (long)blockIdx.x * blockDim.x + threadIdx.x;
    long stride = (long)gridDim.x * blockDim.x;
    for (; i < n; i += stride) p[i] = 0.0f;
}

// ------------------------------------------------- edge scatter (segment_sum)
// One thread handles 4 consecutive features of one edge (float4 gather,
// 4x global_atomic_add_f32 scatter). DO_CNT also accumulates the degree.
template<int DIM, bool DO_CNT>
__global__ void k_scatter(const float* __restrict__ x,
                          const long long* __restrict__ src,
                          const long long* __restrict__ dst,
                          float* __restrict__ agg,
                          float* __restrict__ cnt) {
    constexpr int G = DIM / 4;                 // thread-groups per edge
    long tid = (long)blockIdx.x * blockDim.x + threadIdx.x;
    long e = tid / G;
    int  g = (int)(tid % G);
    if (e >= (long)NEDGES) return;
    int s = (int)src[e];
    int d = (int)dst[e];
    float4 v = reinterpret_cast<const float4*>(x + (long)s * DIM)[g];
    float* o = agg + (long)d * DIM + g * 4;
    __hip_atomic_fetch_add(o + 0, v.x, __ATOMIC_RELAXED, __HIP_MEMORY_SCOPE_AGENT);
    __hip_atomic_fetch_add(o + 1, v.y, __ATOMIC_RELAXED, __HIP_MEMORY_SCOPE_AGENT);
    __hip_atomic_fetch_add(o + 2, v.z, __ATOMIC_RELAXED, __HIP_MEMORY_SCOPE_AGENT);
    __hip_atomic_fetch_add(o + 3, v.w, __ATOMIC_RELAXED, __HIP_MEMORY_SCOPE_AGENT);
    if (DO_CNT && g == 0)
        __hip_atomic_fetch_add(cnt + d, 1.0f, __ATOMIC_RELAXED, __HIP_MEMORY_SCOPE_AGENT);
}

// -------------------------------------------------- fused SAGE layer (WMMA)
// out[M x NO] = act( (agg/cnt) @ Wl + bias + xin @ Wr )
// Block: (NO/16) waves; each wave owns one 16x16 output tile of a 16-row slab.
// V_WMMA_F32_16X16X4_F32 fragment mapping (ISA 7.12.2):
//   A (16x4 f32):  lane 0-15 -> row=lane,  v0=K+0, v1=K+1
//                  lane16-31 -> row=lane-16, v0=K+2, v1=K+3
//   B (4x16 f32):  mirrored (col = lane&15, same K split)
//   C/D (16x16):   VGPR r: lanes 0-15 -> M=r, lanes 16-31 -> M=r+8; N=lane&15
template<int K, int NO, bool RELU>
__global__ void k_sage_gemm(const float* __restrict__ agg,
                            const float* __restrict__ cnt,
                            const float* __restrict__ xin,
                            const float* __restrict__ Wl,
                            const float* __restrict__ bias,
                            const float* __restrict__ Wr,
                            float* __restrict__ out) {
    const int wave  = threadIdx.x >> 5;        // N-tile index
    const int lane  = threadIdx.x & 31;
    const int r16   = lane & 15;
    const int khalf = lane >> 4;               // 0: K+0/1, 1: K+2/3
    const int mbase = blockIdx.x * 16;
    const int nbase = wave * 16;
    const int arow  = mbase + r16;             // row this lane feeds into A frags
    const int ncol  = nbase + r16;             // col this lane feeds into B frags

    const float rcp = 1.0f / fmaxf(cnt[arow], 1.0f);
    const float* aggRow = agg + (long)arow * K;
    const float* xRow   = xin + (long)arow * K;

    v8f acc = {};
#pragma unroll
    for (int k0 = 0; k0 < K; k0 += 4) {
        const int kk = k0 + khalf * 2;
        v2f am, ax, bl, br;
        am.x = aggRow[kk]     * rcp;
        am.y = aggRow[kk + 1] * rcp;
        ax.x = xRow[kk];
        ax.y = xRow[kk + 1];
        bl.x = Wl[(long)kk * NO + ncol];
        bl.y = Wl[(long)(kk + 1) * NO + ncol];
        br.x = Wr[(long)kk * NO + ncol];
        br.y = Wr[(long)(kk + 1) * NO + ncol];
        acc = __builtin_amdgcn_wmma_f32_16x16x4_f32(
                  false, am, false, bl, (short)0, acc, false, false);
        acc = __builtin_amdgcn_wmma_f32_16x16x4_f32(
                  false, ax, false, br, (short)0, acc, false, false);
    }

    const float b = bias[ncol];
#pragma unroll
    for (int r = 0; r < 8; ++r) {
        const int orow = mbase + r + khalf * 8;
        float v = acc[r] + b;
        if (RELU) v = fmaxf(v, 0.0f);
        out[(long)orow * NO + ncol] = v;
    }
}

// ---------------------------------------------------------------- launcher
extern "C" void kernel_launch(void* const* d_in, const int* in_sizes, int n_in,
                              void* d_out, int out_size, void* d_ws, size_t ws_size,
                              hipStream_t stream) {
    const float*     x   = (const float*)d_in[0];
    const long long* ei  = (const long long*)d_in[1];   // int64 in reference
    const float*     Wl1 = (const float*)d_in[2];
    const float*     bl1 = (const float*)d_in[3];
    const float*     Wr1 = (const float*)d_in[4];
    const float*     Wl2 = (const float*)d_in[5];
    const float*     bl2 = (const float*)d_in[6];
    const float*     Wr2 = (const float*)d_in[7];
    float* out = (float*)d_out;

    // workspace layout: [agg1 | cnt | agg2 | h]  (first three zeroed each call)
    float* agg1 = (float*)d_ws;
    float* cnt  = agg1 + (size_t)NNODES * DIN;
    float* agg2 = cnt  + (size_t)NNODES;
    float* h    = agg2 + (size_t)NNODES * DHID;

    const long long* src = ei;
    const long long* dst = ei + NEDGES;

    // 1) zero accumulators (agg1, cnt, agg2 are contiguous)
    const long nz = (long)NNODES * (DIN + 1 + DHID);
    k_zero<<<4096, 256, 0, stream>>>(agg1, nz);

    // 2) layer-1 scatter: E * (64/4) threads
    {
        long threads = (long)NEDGES * (DIN / 4);
        int blocks = (int)((threads + 255) / 256);
        k_scatter<DIN, true><<<blocks, 256, 0, stream>>>(x, src, dst, agg1, cnt);
    }

    // 3) layer-1 fused GEMM + bias + relu -> h   (50000 rows = 3125 tile-rows)
    k_sage_gemm<DIN, DHID, true><<<NNODES / 16, (DHID / 16) * 32, 0, stream>>>(
        agg1, cnt, x, Wl1, bl1, Wr1, h);

    // 4) layer-2 scatter over h: E * (128/4) threads (reuses cnt)
    {
        long threads = (long)NEDGES * (DHID / 4);
        int blocks = (int)((threads + 255) / 256);
        k_scatter<DHID, false><<<blocks, 256, 0, stream>>>(h, src, dst, agg2, cnt);
    }

    // 5) layer-2 fused GEMM + bias -> out
    k_sage_gemm<DHID, DOUT, false><<<NNODES / 16, (DOUT / 16) * 32, 0, stream>>>(
        agg2, cnt, h, Wl2, bl2, Wr2, out);
}